// DelayAudio_14998025798077
// MI455X (gfx1250) — compile-verified
//
#include <hip/hip_runtime.h>

#define TILE      2048
#define NTHREADS  256
#define ELEMS_PT  (TILE / NTHREADS)     // 8 outputs per thread
#define CCH       8                      // CODEC_CHANNELS (codec constant)
#define SOS_F     1024.0f
#define EOS_F     1025.0f
#define PAD_F     1026.0f

// Native vector types (HIP's float4/int4 are classes; builtins need these)
typedef float __attribute__((ext_vector_type(4))) f32x4;
typedef int   __attribute__((ext_vector_type(4))) i32x4;

// Address-space-typed int pointers for the async-to-LDS builtin
typedef __attribute__((address_space(1))) int gbl_int;
typedef __attribute__((address_space(3))) int lds_int;

__global__ __launch_bounds__(NTHREADS)
void delay_audio_kernel(const int* __restrict__ audio,     // [B*CCH*T] int32 tokens
                        const int* __restrict__ audio_len, // [B]
                        const int* __restrict__ target,    // [1]
                        float*     __restrict__ out,       // [B*CCH*Tout]
                        float*     __restrict__ out_len,   // [B]
                        int B, int T)
{
    __shared__ int lds[TILE];

    const int Tout = T + CCH;
    const int row  = blockIdx.y;        // b*CCH + c
    const int b    = row >> 3;          // CCH == 8
    const int c    = row & 7;
    const int t0   = blockIdx.x * TILE;
    const int tid  = threadIdx.x;

    const int offset = (target[0] != 0) ? 0 : 1;
    const int start  = c + offset;
    const int len    = audio_len[b];

    const int* arow = audio + (long long)row * T;

    // ---- Stage the shifted audio window [t0-start, t0-start+TILE) into LDS
    // via async global->LDS DMA. Clamp indices into [0, T-1]: out-of-window
    // lanes fetch valid garbage that is masked by the selects below.
    const int src0 = t0 - start;
#pragma unroll
    for (int k = 0; k < ELEMS_PT; ++k) {
        const int j   = tid + k * NTHREADS;              // contiguous 128B per wave per op
        int       src = src0 + j;
        src = (src < 0) ? 0 : ((src >= T) ? (T - 1) : src);
#if __has_builtin(__builtin_amdgcn_global_load_async_to_lds_b32)
        __builtin_amdgcn_global_load_async_to_lds_b32(
            (gbl_int*)(arow + src), (lds_int*)(&lds[j]), 0, 0);
#else
        lds[j] = arow[src];
#endif
    }
#if __has_builtin(__builtin_amdgcn_s_wait_asynccnt)
    __builtin_amdgcn_s_wait_asynccnt(0);   // this wave's async LDS writes done
#endif
    __syncthreads();                       // cross-wave visibility

    // ---- Convert + 3-way select (reference order: SOS, then EOS, then PAD
    // overriding), two aligned b128 NT stores per thread.
    const int eos_t = start + len;         // t >= eos_t -> EOS
    const int pad_t = len + CCH;           // t >= pad_t -> PAD (overrides EOS)
    float* orow = out + (long long)row * Tout;

    const int t = t0 + tid * ELEMS_PT;
    if (t < Tout) {
        const int j0 = tid * ELEMS_PT;
        float r[ELEMS_PT];
#pragma unroll
        for (int q = 0; q < ELEMS_PT / 4; ++q) {
            const i32x4 v = *(const i32x4*)&lds[j0 + 4 * q];
            r[4*q+0] = (float)v.x; r[4*q+1] = (float)v.y;
            r[4*q+2] = (float)v.z; r[4*q+3] = (float)v.w;
        }
#pragma unroll
        for (int k = 0; k < ELEMS_PT; ++k) {
            const int tt = t + k;
            float x = r[k];
            if (tt <  start) x = SOS_F;
            if (tt >= eos_t) x = EOS_F;
            if (tt >= pad_t) x = PAD_F;
            r[k] = x;
        }
        if (t + ELEMS_PT - 1 < Tout) {
#pragma unroll
            for (int q = 0; q < ELEMS_PT / 4; ++q) {
                f32x4 s = { r[4*q+0], r[4*q+1], r[4*q+2], r[4*q+3] };
                __builtin_nontemporal_store(s, (f32x4*)&orow[t + 4*q]);
            }
        } else {
            for (int k = 0; k < ELEMS_PT && (t + k) < Tout; ++k)
                __builtin_nontemporal_store(r[k], &orow[t + k]);
        }
    }

    // ---- Second tuple output: audio_len + CCH  (tiny, done by one block)
    if (blockIdx.x == 0 && blockIdx.y == 0 && tid < B) {
        out_len[tid] = (float)(audio_len[tid] + CCH);
    }
}

extern "C" void kernel_launch(void* const* d_in, const int* in_sizes, int n_in,
                              void* d_out, int out_size, void* d_ws, size_t ws_size,
                              hipStream_t stream)
{
    const int* audio     = (const int*)d_in[0];
    const int* audio_len = (const int*)d_in[1];
    const int* target    = (const int*)d_in[2];
    float*     out       = (float*)d_out;

    const int B = in_sizes[1];             // 64
    const int T = in_sizes[0] / (B * CCH); // 32768
    const int Tout = T + CCH;

    float* out_len = out + (size_t)B * CCH * Tout;

    dim3 grid((Tout + TILE - 1) / TILE, (unsigned)(B * CCH));
    delay_audio_kernel<<<grid, NTHREADS, 0, stream>>>(
        audio, audio_len, target, out, out_len, B, T);
}